// SVDHead_81836306858176
// MI455X (gfx1250) — compile-verified
//
#include <hip/hip_runtime.h>

// ---------------------------------------------------------------------------
// SVDHead for MI455X (gfx1250, wave32).
//
// Stage 1 (bandwidth-bound): H[b] = src^T diag(mask_tgt) tgt via
//   V_WMMA_F32_16X16X4_F32, 4 points per WMMA, operands staged through LDS.
// Stage 2 (negligible): per-batch Kabsch rotation via Horn quaternion
//   (4x4 Jacobi eigensolver), equivalent to V diag(1,1,sign) U^T.
// ---------------------------------------------------------------------------

typedef float v2f __attribute__((ext_vector_type(2)));
typedef float v8f __attribute__((ext_vector_type(8)));

#define NPTS    2048
#define THREADS 256
#define WAVES   8      // 256 / wave32
#define CHUNKS  8      // 8 chunks * 32 pts * 8 waves = 2048 pts per block

__global__ __launch_bounds__(THREADS)
void svd_head_kernel(const float* __restrict__ src,
                     const float* __restrict__ tgt,
                     const float* __restrict__ mask_tgt,
                     float* __restrict__ out)
{
    __shared__ float sStage[WAVES][96];   // 32 pts * 3 comps per wave
    __shared__ float tStage[WAVES][96];   // masked tgt
    __shared__ float hSum[9];             // H accumulated across waves

    const int b    = blockIdx.x;
    const int tid  = threadIdx.x;
    const int lane = tid & 31;
    const int wave = tid >> 5;

    if (tid < 9) hSum[tid] = 0.0f;
    __syncthreads();

    // WMMA f32 16x16x4 operand addressing (ISA 7.12.2):
    //   A 16x4 : lane&15 = M row, K = 2*(lane>>4) + vgpr_index
    //   B 4x16 : lane&15 = N col, same K mapping (symmetric layout)
    // -> both A and B read stage[(4g + 2h)*3 + comp] and stage[idx+3].
    // Rows/cols 3..15 of the 16x16 tile are don't-care: clamp comp to 2.
    const int r     = lane & 15;
    const int half  = lane >> 4;
    const int comp  = (r < 3) ? r : 2;
    const int cbase = 6 * half + comp;

    const size_t base = (size_t)b * NPTS;
    const float* __restrict__ srcB = src + base * 3;
    const float* __restrict__ tgtB = tgt + base * 3;
    const float* __restrict__ mB   = mask_tgt + base;

    float* sW = &sStage[wave][0];
    float* tW = &tStage[wave][0];

    v8f acc = {0.f, 0.f, 0.f, 0.f, 0.f, 0.f, 0.f, 0.f};

    for (int c = 0; c < CHUNKS; ++c) {
        const int n = wave * (CHUNKS * 32) + c * 32 + lane;

        // prefetch next chunk while this one is consumed (global_prefetch_b8)
        if (c + 1 < CHUNKS) {
            __builtin_prefetch(srcB + (size_t)(n + 32) * 3, 0, 1);
            __builtin_prefetch(tgtB + (size_t)(n + 32) * 3, 0, 1);
        }

        const float m  = mB[n];
        const float* sp = srcB + 3 * (size_t)n;
        const float* tp = tgtB + 3 * (size_t)n;
        const float sx = sp[0], sy = sp[1], sz = sp[2];
        const float tx = tp[0] * m, ty = tp[1] * m, tz = tp[2] * m;

        // stage 32 points, [point][comp] layout
        sW[lane * 3 + 0] = sx; sW[lane * 3 + 1] = sy; sW[lane * 3 + 2] = sz;
        tW[lane * 3 + 0] = tx; tW[lane * 3 + 1] = ty; tW[lane * 3 + 2] = tz;

        // same-wave LDS producer->consumer: LDS is in-order per wave;
        // wait the split DS counter and fence the compiler.
        asm volatile("s_wait_dscnt 0" ::: "memory");
        __builtin_amdgcn_wave_barrier();

        #pragma unroll
        for (int g = 0; g < 8; ++g) {      // 8 WMMAs x K=4 points = 32 points
            const int idx = g * 12 + cbase;
            v2f a, bm;
            a.x  = sW[idx];     a.y  = sW[idx + 3];   // ds_load_2addr pair
            bm.x = tW[idx];     bm.y = tW[idx + 3];
            acc = __builtin_amdgcn_wmma_f32_16x16x4_f32(
                      false, a, false, bm, (short)0, acc, false, false);
        }
        __builtin_amdgcn_wave_barrier();
    }

    // C/D layout: VGPR c, lanes 0..15 -> D[c][lane]. H lives in acc[0..2],
    // lanes 0..2. Reduce the 8 waves through LDS float atomics.
    if (lane < 3) {
        atomicAdd(&hSum[0 * 3 + lane], acc[0]);
        atomicAdd(&hSum[1 * 3 + lane], acc[1]);
        atomicAdd(&hSum[2 * 3 + lane], acc[2]);
    }
    __syncthreads();

    // ------------------- Stage 2: Kabsch via Horn quaternion ---------------
    if (tid == 0) {
        const float Sxx = hSum[0], Sxy = hSum[1], Sxz = hSum[2];
        const float Syx = hSum[3], Syy = hSum[4], Syz = hSum[5];
        const float Szx = hSum[6], Szy = hSum[7], Szz = hSum[8];

        float A[4][4];
        A[0][0] =  Sxx + Syy + Szz;
        A[0][1] =  Syz - Szy;  A[0][2] = Szx - Sxz;  A[0][3] = Sxy - Syx;
        A[1][1] =  Sxx - Syy - Szz;  A[1][2] = Sxy + Syx;  A[1][3] = Szx + Sxz;
        A[2][2] = -Sxx + Syy - Szz;  A[2][3] = Syz + Szy;
        A[3][3] = -Sxx - Syy + Szz;
        A[1][0] = A[0][1]; A[2][0] = A[0][2]; A[3][0] = A[0][3];
        A[2][1] = A[1][2]; A[3][1] = A[1][3]; A[3][2] = A[2][3];

        float V[4][4] = {{1,0,0,0},{0,1,0,0},{0,0,1,0},{0,0,0,1}};

        const int PP[6] = {0,0,0,1,1,2};
        const int QQ[6] = {1,2,3,2,3,3};

        for (int sweep = 0; sweep < 8; ++sweep) {
            #pragma unroll
            for (int e = 0; e < 6; ++e) {
                const int p = PP[e], q = QQ[e];
                const float apq = A[p][q];
                if (__builtin_fabsf(apq) > 1e-30f) {
                    const float tau = (A[q][q] - A[p][p]) / (2.0f * apq);
                    const float t   = __builtin_copysignf(1.0f, tau) /
                                      (__builtin_fabsf(tau) + __builtin_sqrtf(1.0f + tau * tau));
                    const float cs  = 1.0f / __builtin_sqrtf(1.0f + t * t);
                    const float sn  = t * cs;
                    const float app = A[p][p], aqq = A[q][q];
                    A[p][p] = app - t * apq;
                    A[q][q] = aqq + t * apq;
                    A[p][q] = 0.0f; A[q][p] = 0.0f;
                    #pragma unroll
                    for (int rr = 0; rr < 4; ++rr) {
                        if (rr != p && rr != q) {
                            const float arp = A[rr][p], arq = A[rr][q];
                            A[rr][p] = cs * arp - sn * arq;  A[p][rr] = A[rr][p];
                            A[rr][q] = sn * arp + cs * arq;  A[q][rr] = A[rr][q];
                        }
                    }
                    #pragma unroll
                    for (int rr = 0; rr < 4; ++rr) {
                        const float vrp = V[rr][p], vrq = V[rr][q];
                        V[rr][p] = cs * vrp - sn * vrq;
                        V[rr][q] = sn * vrp + cs * vrq;
                    }
                }
            }
        }

        // pick eigenvector of the largest eigenvalue (constant-indexed selects)
        float bd = A[0][0];
        float qw = V[0][0], qx = V[1][0], qy = V[2][0], qz = V[3][0];
        if (A[1][1] > bd) { bd = A[1][1]; qw = V[0][1]; qx = V[1][1]; qy = V[2][1]; qz = V[3][1]; }
        if (A[2][2] > bd) { bd = A[2][2]; qw = V[0][2]; qx = V[1][2]; qy = V[2][2]; qz = V[3][2]; }
        if (A[3][3] > bd) { bd = A[3][3]; qw = V[0][3]; qx = V[1][3]; qy = V[2][3]; qz = V[3][3]; }

        const float inv = 1.0f / __builtin_sqrtf(qw*qw + qx*qx + qy*qy + qz*qz + 1e-30f);
        qw *= inv; qx *= inv; qy *= inv; qz *= inv;

        float* o = out + (size_t)b * 9;
        o[0] = 1.0f - 2.0f * (qy*qy + qz*qz);
        o[1] = 2.0f * (qx*qy - qw*qz);
        o[2] = 2.0f * (qx*qz + qw*qy);
        o[3] = 2.0f * (qx*qy + qw*qz);
        o[4] = 1.0f - 2.0f * (qx*qx + qz*qz);
        o[5] = 2.0f * (qy*qz - qw*qx);
        o[6] = 2.0f * (qx*qz - qw*qy);
        o[7] = 2.0f * (qy*qz + qw*qx);
        o[8] = 1.0f - 2.0f * (qx*qx + qy*qy);
    }
}

extern "C" void kernel_launch(void* const* d_in, const int* in_sizes, int n_in,
                              void* d_out, int out_size, void* d_ws, size_t ws_size,
                              hipStream_t stream) {
    (void)n_in; (void)out_size; (void)d_ws; (void)ws_size;
    const float* src      = (const float*)d_in[0];
    const float* tgt      = (const float*)d_in[1];
    // d_in[2] = kpt_src_mask is unused by the reference computation
    const float* mask_tgt = (const float*)d_in[3];
    float* out = (float*)d_out;

    const int B = in_sizes[0] / (NPTS * 3);   // 4096
    svd_head_kernel<<<dim3(B), dim3(THREADS), 0, stream>>>(src, tgt, mask_tgt, out);
}